// bilinear_Net_3_84954453115059
// MI455X (gfx1250) — compile-verified
//
#include <hip/hip_runtime.h>
#include <hip/hip_bf16.h>

// ---------------------------------------------------------------------------
// MI455X (gfx1250, wave32) implementation of the bilinear GNN reference.
//
// Key transformation: GCN aggregation is linear, and the deg==0 fallback
// commutes through the per-node linear layer:
//     relu(agg(h) @ W^T + b) == relu( (deg>0 ? segsum(h@W^T)/deg : h@W^T) + b )
// so we run the dense transform FIRST (WMMA fp32) and scatter the *smaller*
// transformed features over the edges (100 floats/edge for layer 1 instead of
// 128, 20 instead of 100 for layer 2). Edge traffic ~0.77 GB + 0.13 GB; the
// 40 MB scatter accumulator lives in the 192 MB L2 so the f32 atomics resolve
// on-chip.
//
// Matrix ops use V_WMMA_F32_16X16X4_F32 (fp32 A/B/C) to preserve reference
// precision — the GEMMs are <3 GFLOP total, far below any WMMA ceiling.
// ---------------------------------------------------------------------------

typedef __attribute__((ext_vector_type(2))) float v2f;
typedef __attribute__((ext_vector_type(8))) float v8f;

#define DIN    128
#define D1     100
#define D2     20
#define DSF    64
#define H1C    32
#define H2C    8
#define NBATCH 256

// ---------------------------------------------------------------------------
// Generic NT GEMM:  C[M,Nd] = A[M,Kd] . W[Nd,Kd]^T  (+ bias[Nd] if non-null)
// Requirements: M % 16 == 0, Kd % 4 == 0 (all call sites satisfy both).
//
// One wave owns a 16-row M-tile and NT accumulator tiles (16 cols each); A is
// streamed exactly once per row-block, W stays L2-resident.
//
// No branches in the K loop:
//  * out-of-range B columns are CLAMPED (not masked): column n of D depends
//    only on column n of B, and we never store columns >= Nd, so garbage from
//    a clamped (in-bounds) address is harmless.
//  * each lane's two K values are memory-adjacent -> one b64 load for A and
//    one per B tile.
//
// VGPR layouts per CDNA5 ISA 7.12.2 (wave32), symmetric for A and B:
//   A 16x4 f32 : VGPR0 = {K=k0 (lanes 0-15) | K=k0+2 (lanes 16-31)},
//                VGPR1 = {K=k0+1            | K=k0+3}
//   B 4x16 f32 : same half-wave split over K rows, N across lanes.
//   C/D 16x16  : VGPR r, lane<16 -> (M=r, N=lane); lane>=16 -> (M=r+8, N=lane-16)
// ---------------------------------------------------------------------------
template <int NT>
__global__ void k_gemm_nt_wmma(const float* __restrict__ A,
                               const float* __restrict__ W,
                               const float* __restrict__ bias,
                               float* __restrict__ C,
                               int M, int Nd, int Kd) {
    const int wave = (int)((blockIdx.x * blockDim.x + threadIdx.x) >> 5);
    const int lane = (int)(threadIdx.x & 31);
    const int mtiles = M >> 4;            // M % 16 == 0
    if (wave >= mtiles) return;

    const int m0   = wave << 4;
    const int ln   = lane & 15;
    const int half = lane >> 4;
    const int mrow = m0 + ln;

    // Per-lane base pointers; K offset advances by 4 floats per step.
    const float* ap = A + (long long)mrow * Kd + 2 * half;
    const float* wp[NT];
#pragma unroll
    for (int tn = 0; tn < NT; ++tn) {
        int ncol = tn * 16 + ln;
        if (ncol >= Nd) ncol = Nd - 1;    // clamp: see header comment
        wp[tn] = W + (long long)ncol * Kd + 2 * half;
    }

    v8f acc[NT];
#pragma unroll
    for (int i = 0; i < NT; ++i) acc[i] = (v8f){};

    for (int k0 = 0; k0 < Kd; k0 += 4) {
        const v2f a = *(const v2f*)(ap + k0);           // 8B-aligned b64 load
#pragma unroll
        for (int tn = 0; tn < NT; ++tn) {
            const v2f b = *(const v2f*)(wp[tn] + k0);   // 8B-aligned b64 load
            acc[tn] = __builtin_amdgcn_wmma_f32_16x16x4_f32(
                false, a, false, b, (short)0, acc[tn], false, false);
        }
    }

#pragma unroll
    for (int tn = 0; tn < NT; ++tn) {
        const int cn = tn * 16 + ln;
        if (cn >= Nd) continue;
        const float bv = bias ? bias[cn] : 0.0f;
#pragma unroll
        for (int r = 0; r < 8; ++r) {
            const int cm = m0 + r + half * 8;           // always < M (M%16==0)
            C[(long long)cm * Nd + cn] = acc[tn][r] + bv;
        }
    }
}

// ---------------------------------------------------------------------------
// Edge scatter: acc[dst[e]][:] += Y[src[e]][:]. Thread = (edge, 16B chunk).
// Row strides 100*4=400B and 20*4=80B are 16B aligned -> float4 loads.
// Accumulator is L2-resident (40 MB << 192 MB L2).
// ---------------------------------------------------------------------------
__global__ void k_edge_scatter(const float* __restrict__ Y,
                               const int* __restrict__ src,
                               const int* __restrict__ dst,
                               float* __restrict__ acc,
                               int E, int d, int nchunk) {
    const long long t = (long long)blockIdx.x * blockDim.x + threadIdx.x;
    if (t >= (long long)E * nchunk) return;
    const int e = (int)(t / nchunk);
    const int c = (int)(t % nchunk);
    const int s = src[e];
    const int dn = dst[e];
    const int off = c * 4;
    const float* yp = Y + (long long)s * d + off;
    float* ap = acc + (long long)dn * d + off;
    const float4 v = *(const float4*)yp;                // d % 4 == 0 here
    atomicAdd(ap + 0, v.x);
    atomicAdd(ap + 1, v.y);
    atomicAdd(ap + 2, v.z);
    atomicAdd(ap + 3, v.w);
}

__global__ void k_degree(const int* __restrict__ dst, float* __restrict__ deg, int E) {
    const int t = blockIdx.x * blockDim.x + threadIdx.x;
    if (t < E) atomicAdd(&deg[dst[t]], 1.0f);
}

__global__ void k_gcnt(const int* __restrict__ n2g, float* __restrict__ gcnt, int N) {
    const int t = blockIdx.x * blockDim.x + threadIdx.x;
    if (t < N) atomicAdd(&gcnt[n2g[t]], 1.0f);
}

// Per-node finish: z = deg>0 ? acc/deg : y_self; h = relu(z + bias). H may
// alias acc (pure elementwise).
__global__ void k_finish(const float* __restrict__ acc,
                         const float* __restrict__ Y,
                         const float* __restrict__ deg,
                         const float* __restrict__ bias,
                         float* __restrict__ H, int N, int d) {
    const long long t = (long long)blockIdx.x * blockDim.x + threadIdx.x;
    if (t >= (long long)N * d) return;
    const int n = (int)(t / d);
    const int c = (int)(t % d);
    const float dv = deg[n];
    const float z = (dv > 0.0f) ? (acc[t] / dv) : Y[t];
    H[t] = fmaxf(z + bias[c], 0.0f);
}

__global__ void k_graph_sum(const float* __restrict__ h2,
                            const int* __restrict__ n2g,
                            float* __restrict__ gsum, int N) {
    const int t = blockIdx.x * blockDim.x + threadIdx.x;
    if (t >= N * D2) return;
    const int n = t / D2, c = t % D2;
    atomicAdd(&gsum[n2g[n] * D2 + c], h2[t]);
}

__global__ void k_graph_mean(const float* __restrict__ gsum,
                             const float* __restrict__ gcnt,
                             float* __restrict__ hg) {
    const int t = blockIdx.x * blockDim.x + threadIdx.x;
    if (t >= NBATCH * D2) return;
    hg[t] = gsum[t] / fmaxf(gcnt[t / D2], 1.0f);
}

// x[b,j] = (hg[b,:] @ Wbil[:,j]) * self_feat[b,j]     (256x20x64 MACs, tiny)
__global__ void k_bil_fuse(const float* __restrict__ hg,
                           const float* __restrict__ Wbil,
                           const float* __restrict__ sf,
                           float* __restrict__ x) {
    const int t = blockIdx.x * blockDim.x + threadIdx.x;
    if (t >= NBATCH * DSF) return;
    const int b = t >> 6, j = t & 63;
    float s = 0.0f;
#pragma unroll
    for (int k = 0; k < D2; ++k) s += hg[b * D2 + k] * Wbil[k * DSF + j];
    x[t] = s * sf[t];
}

// Training-mode batchnorm + relu, two-pass (matches E[(x-mu)^2] form exactly).
// grid = C blocks, blockDim = B = 256 (power of 2). In-place safe.
__global__ void k_bn_relu(float* __restrict__ X,
                          const float* __restrict__ g,
                          const float* __restrict__ beta, int C) {
    const int c = blockIdx.x;
    const int r = threadIdx.x;           // B == blockDim.x == 256
    __shared__ float sm[256];
    const float v = X[r * C + c];
    sm[r] = v;
    __syncthreads();
    for (int s = 128; s > 0; s >>= 1) {
        if (r < s) sm[r] += sm[r + s];
        __syncthreads();
    }
    const float mu = sm[0] * (1.0f / 256.0f);
    __syncthreads();
    const float d = v - mu;
    sm[r] = d * d;
    __syncthreads();
    for (int s = 128; s > 0; s >>= 1) {
        if (r < s) sm[r] += sm[r + s];
        __syncthreads();
    }
    const float var = sm[0] * (1.0f / 256.0f);
    X[r * C + c] = fmaxf(g[c] * d * rsqrtf(var + 1e-5f) + beta[c], 0.0f);
}

// ---------------------------------------------------------------------------
extern "C" void kernel_launch(void* const* d_in, const int* in_sizes, int n_in,
                              void* d_out, int out_size, void* d_ws, size_t ws_size,
                              hipStream_t stream) {
    const float* feat   = (const float*)d_in[0];
    const float* sfeat  = (const float*)d_in[1];
    const int*   src    = (const int*)d_in[2];
    const int*   dst    = (const int*)d_in[3];
    const int*   n2g    = (const int*)d_in[4];
    const float* W1     = (const float*)d_in[5];
    const float* b1     = (const float*)d_in[6];
    const float* W2     = (const float*)d_in[7];
    const float* b2     = (const float*)d_in[8];
    const float* Wbil   = (const float*)d_in[9];
    const float* fc1_w  = (const float*)d_in[10];
    const float* fc1_b  = (const float*)d_in[11];
    const float* fc2_w  = (const float*)d_in[12];
    const float* fc2_b  = (const float*)d_in[13];
    const float* fc3_w  = (const float*)d_in[14];
    const float* fc3_b  = (const float*)d_in[15];
    const float* g1     = (const float*)d_in[16];
    const float* beta1  = (const float*)d_in[17];
    const float* g2     = (const float*)d_in[18];
    const float* beta2  = (const float*)d_in[19];
    float* out = (float*)d_out;

    const int N = in_sizes[0] / DIN;     // 100000
    const int E = in_sizes[2];           // 1600000
    const int B = NBATCH;                // 256

    // ---- workspace layout (floats), ~92 MB total ----
    float* ws = (float*)d_ws;
    long long o = 0;
    float* Y1   = ws + o; o += (long long)N * D1;          // pre-agg layer1
    float* Y2   = ws + o; o += (long long)N * D2;          // pre-agg layer2
    float* acc1 = ws + o; const long long z0 = o; o += (long long)N * D1; // -> h1 in place
    float* acc2 = ws + o; o += (long long)N * D2;          // -> h2 in place
    float* deg  = ws + o; o += ((long long)N + 63) & ~63LL;
    float* gsum = ws + o; o += (long long)B * D2;
    float* gcnt = ws + o; const long long z1 = o + B; o += B;
    float* hg   = ws + o; o += (long long)B * D2;
    float* xbuf = ws + o; o += (long long)B * DSF;
    float* t1   = ws + o; o += (long long)B * H1C;
    float* t2   = ws + o; o += (long long)B * H2C;
    (void)ws_size; (void)n_in; (void)out_size;

    // zero the atomic accumulators (one contiguous span: acc1..gcnt)
    (void)hipMemsetAsync(ws + z0, 0, (size_t)(z1 - z0) * sizeof(float), stream);

    const int TB = 256;
    // ---- layer 1: Y1 = feat @ W1^T  (M=N, Nd=100 -> 7 tiles, Kd=128) ----
    {
        const int mtiles = N / 16;
        k_gemm_nt_wmma<7><<<(mtiles * 32 + TB - 1) / TB, TB, 0, stream>>>(
            feat, W1, nullptr, Y1, N, D1, DIN);
    }
    k_degree<<<(E + TB - 1) / TB, TB, 0, stream>>>(dst, deg, E);
    k_gcnt<<<(N + TB - 1) / TB, TB, 0, stream>>>(n2g, gcnt, N);

    // ---- scatter Y1 over edges, finish -> h1 (in acc1) ----
    {
        const long long tot = (long long)E * (D1 / 4);
        k_edge_scatter<<<(int)((tot + TB - 1) / TB), TB, 0, stream>>>(
            Y1, src, dst, acc1, E, D1, D1 / 4);
        const long long nf = (long long)N * D1;
        k_finish<<<(int)((nf + TB - 1) / TB), TB, 0, stream>>>(
            acc1, Y1, deg, b1, acc1, N, D1);
    }

    // ---- layer 2: Y2 = h1 @ W2^T  (Nd=20 -> 2 tiles, Kd=100) ----
    {
        const int mtiles = N / 16;
        k_gemm_nt_wmma<2><<<(mtiles * 32 + TB - 1) / TB, TB, 0, stream>>>(
            acc1, W2, nullptr, Y2, N, D2, D1);
        const long long tot = (long long)E * (D2 / 4);
        k_edge_scatter<<<(int)((tot + TB - 1) / TB), TB, 0, stream>>>(
            Y2, src, dst, acc2, E, D2, D2 / 4);
        const long long nf = (long long)N * D2;
        k_finish<<<(int)((nf + TB - 1) / TB), TB, 0, stream>>>(
            acc2, Y2, deg, b2, acc2, N, D2);
    }

    // ---- per-graph mean ----
    k_graph_sum<<<(N * D2 + TB - 1) / TB, TB, 0, stream>>>(acc2, n2g, gsum, N);
    k_graph_mean<<<(B * D2 + TB - 1) / TB, TB, 0, stream>>>(gsum, gcnt, hg);

    // ---- bilinear fuse + head MLP ----
    k_bil_fuse<<<(B * DSF + TB - 1) / TB, TB, 0, stream>>>(hg, Wbil, sfeat, xbuf);

    {   // fc1: [256,64] @ [32,64]^T -> [256,32]
        k_gemm_nt_wmma<2><<<(B / 16 * 32 + TB - 1) / TB, TB, 0, stream>>>(
            xbuf, fc1_w, fc1_b, t1, B, H1C, DSF);
    }
    k_bn_relu<<<H1C, 256, 0, stream>>>(t1, g1, beta1, H1C);

    {   // fc2: [256,32] @ [8,32]^T -> [256,8]
        k_gemm_nt_wmma<1><<<(B / 16 * 32 + TB - 1) / TB, TB, 0, stream>>>(
            t1, fc2_w, fc2_b, t2, B, H2C, H1C);
    }
    k_bn_relu<<<H2C, 256, 0, stream>>>(t2, g2, beta2, H2C);

    {   // fc3: [256,8] @ [1,8]^T -> [256,1]  (writes d_out directly)
        k_gemm_nt_wmma<1><<<(B / 16 * 32 + TB - 1) / TB, TB, 0, stream>>>(
            t2, fc3_w, fc3_b, out, B, 1, H2C);
    }
}